// Attention_54305566490745
// MI455X (gfx1250) — compile-verified
//
#include <hip/hip_runtime.h>
#include <math.h>

// Problem constants (from setup_inputs): x(64,512,32,32), q(1,8,1,64),
// Wkv(512,1024), bkv(1024), NH=8.
#define BATCH 64
#define DMODEL 512
#define NTOK 1024         // 32*32
#define NHEADS 8
#define DHEAD 64
#define SCALE 0.125f      // 64^-0.5

typedef __attribute__((ext_vector_type(2))) float v2f;
typedef __attribute__((ext_vector_type(8))) float v8f;

// D(16x16,f32) = A(16x4,f32) x B(4x16,f32) + C   — faithful fp32 matrix path.
__device__ __forceinline__ v8f wmma_f32_4(v2f a, v2f b, v8f c) {
    return __builtin_amdgcn_wmma_f32_16x16x4_f32(
        /*neg_a=*/false, a, /*neg_b=*/false, b,
        /*c_mod=*/(short)0, c, /*reuse_a=*/false, /*reuse_b=*/false);
}

// ---------------------------------------------------------------------------
// K0: fold q into Wk -> wq16[c][16] (cols 8..15 = 0, so WMMA A-fetches are
// branch-free);  cq[h][n] = q_h . (bk_h + pe_n)
// ---------------------------------------------------------------------------
__global__ void attn_prep(const float* __restrict__ q,
                          const float* __restrict__ Wkv,
                          const float* __restrict__ bkv,
                          float* __restrict__ wq16,  // [512][16]
                          float* __restrict__ cq) {  // [8][1024]
    int idx = blockIdx.x * blockDim.x + threadIdx.x;
    if (idx < DMODEL * 16) {
        int c = idx >> 4, h = idx & 15;
        float s = 0.f;
        if (h < NHEADS) {
            #pragma unroll 8
            for (int d = 0; d < DHEAD; ++d)
                s += Wkv[c * (2 * DMODEL) + h * DHEAD + d] * q[h * DHEAD + d];
        }
        wq16[idx] = s;
    } else if (idx < DMODEL * 16 + NHEADS * NTOK) {
        int j = idx - DMODEL * 16;
        int h = j >> 10, n = j & (NTOK - 1);
        const float kLn = 9.210340371976184f / 64.0f;  // ln(10000)/d_model
        float s = 0.f;
        for (int i = 0; i < DHEAD / 2; ++i) {
            float t = __expf(-(float)(2 * i) * kLn);
            float sn, cs; __sincosf((float)n * t, &sn, &cs);
            s += q[h * DHEAD + 2 * i]     * (bkv[h * DHEAD + 2 * i]     + sn);
            s += q[h * DHEAD + 2 * i + 1] * (bkv[h * DHEAD + 2 * i + 1] + cs);
        }
        cq[h * NTOK + n] = s;
    }
}

// ---------------------------------------------------------------------------
// K1: dots[b][h][n] = SCALE * ( sum_c x[b,c,n] * wq[c,h] + cq[h][n] )
// One wave -> 16 n-columns, 8 head rows.  x loads coalesced along n.
// wq16 staged in LDS once per block; dual accumulator chains.
// ---------------------------------------------------------------------------
__global__ void attn_dots(const float* __restrict__ x,
                          const float* __restrict__ wq16,
                          const float* __restrict__ cq,
                          float* __restrict__ dots) {
    __shared__ float swq[DMODEL * 16];   // 32 KB
    const int tid  = threadIdx.x;
    const int wave = tid >> 5;
    const int lane = tid & 31;
    const int half = lane >> 4;          // K pair select
    const int lid  = lane & 15;          // N (token) / M (head) index
    const int b    = blockIdx.y;
    const int n0   = blockIdx.x * 128 + wave * 16;

    for (int i = tid; i < DMODEL * 16; i += 256) swq[i] = wq16[i];
    __syncthreads();

    const float* xb = x + (size_t)b * DMODEL * NTOK + n0 + lid;
    v8f acc0 = {}, acc1 = {};
    #pragma unroll 4
    for (int c0 = 0; c0 < DMODEL; c0 += 8) {
        int k0 = c0 + 2 * half;
        int k1 = c0 + 4 + 2 * half;
        v2f A0, B0, A1, B1;
        // A[m=h, k=c] = wq16[c][h]   (rows 8..15 pre-zeroed)
        A0.x = swq[(k0)     * 16 + lid];
        A0.y = swq[(k0 + 1) * 16 + lid];
        A1.x = swq[(k1)     * 16 + lid];
        A1.y = swq[(k1 + 1) * 16 + lid];
        // B[k=c, n] = x[b, c, n0+n]  — 16 contiguous floats per row
        B0.x = xb[(size_t)(k0)     * NTOK];
        B0.y = xb[(size_t)(k0 + 1) * NTOK];
        B1.x = xb[(size_t)(k1)     * NTOK];
        B1.y = xb[(size_t)(k1 + 1) * NTOK];
        acc0 = wmma_f32_4(A0, B0, acc0);
        acc1 = wmma_f32_4(A1, B1, acc1);
    }
    v8f acc = acc0 + acc1;
    if (half == 0) {                      // rows M=0..7 live in lanes 0..15
        #pragma unroll
        for (int r = 0; r < NHEADS; ++r)
            dots[((size_t)b * NHEADS + r) * NTOK + n0 + lid] =
                (acc[r] + cq[r * NTOK + n0 + lid]) * SCALE;
    }
}

// ---------------------------------------------------------------------------
// K2: softmax over n (1024) per (b,h) row, in place.
// ---------------------------------------------------------------------------
__global__ void attn_softmax(float* __restrict__ att) {
    __shared__ float red[256];
    const int tid = threadIdx.x;
    float* p = att + (size_t)blockIdx.x * NTOK;

    float v[4], m = -3.402823466e38f;
    #pragma unroll
    for (int i = 0; i < 4; ++i) { v[i] = p[tid + i * 256]; m = fmaxf(m, v[i]); }
    red[tid] = m; __syncthreads();
    for (int off = 128; off > 0; off >>= 1) {
        if (tid < off) red[tid] = fmaxf(red[tid], red[tid + off]);
        __syncthreads();
    }
    m = red[0]; __syncthreads();

    float s = 0.f;
    #pragma unroll
    for (int i = 0; i < 4; ++i) { v[i] = __expf(v[i] - m); s += v[i]; }
    red[tid] = s; __syncthreads();
    for (int off = 128; off > 0; off >>= 1) {
        if (tid < off) red[tid] += red[tid + off];
        __syncthreads();
    }
    float inv = 1.f / red[0];
    #pragma unroll
    for (int i = 0; i < 4; ++i) p[tid + i * 256] = v[i] * inv;
}

// ---------------------------------------------------------------------------
// K3: a[b][h][c] = sum_n attn[b][h][n] * x[b][c][n]
// Block: 128 c-rows, full n=1024 reduction (no atomics). x and attn staged
// via LDS (stride 66: even -> b64 loads, still bank-conflict-free).
// ---------------------------------------------------------------------------
#define NC 64
#define TS 66
__global__ void attn_av(const float* __restrict__ x,
                        const float* __restrict__ att,
                        float* __restrict__ a) {
    __shared__ float tile[128 * TS];     // x chunk, 33.8 KB
    __shared__ float at[16 * TS];        // attn chunk, rows 8..15 zero
    const int tid  = threadIdx.x;
    const int wave = tid >> 5;
    const int lane = tid & 31;
    const int half = lane >> 4;
    const int lid  = lane & 15;
    const int b    = blockIdx.y;
    const int cbase = blockIdx.x * 128;          // 4 blocks cover c=0..511

    for (int i = tid; i < 8 * TS; i += 256) at[8 * TS + i] = 0.f;

    const float* xb = x + (size_t)b * DMODEL * NTOK;
    const float* ab = att + (size_t)b * NHEADS * NTOK;
    v8f acc0 = {}, acc1 = {};

    for (int nbase = 0; nbase < NTOK; nbase += NC) {
        // stage x[b, cbase..cbase+127, nbase..nbase+63] — coalesced along n
        for (int i = tid; i < 128 * NC; i += 256) {
            int row = i >> 6, col = i & (NC - 1);
            tile[row * TS + col] = xb[(size_t)(cbase + row) * NTOK + nbase + col];
        }
        // stage attn rows 0..7 for this chunk
        for (int i = tid; i < NHEADS * NC; i += 256) {
            int row = i >> 6, col = i & (NC - 1);
            at[row * TS + col] = ab[row * NTOK + nbase + col];
        }
        __syncthreads();
        #pragma unroll 4
        for (int kk = 0; kk < NC; kk += 8) {
            int k0 = kk + 2 * half;
            int k1 = kk + 4 + 2 * half;
            // A[m=h, k=n] from LDS (b64, 8B-aligned: even stride, even k)
            v2f A0 = *(const v2f*)(at + lid * TS + k0);
            v2f A1 = *(const v2f*)(at + lid * TS + k1);
            // B[k=n, n_col=c] = x[b, c, n] from LDS
            v2f B0 = *(const v2f*)(tile + (wave * 16 + lid) * TS + k0);
            v2f B1 = *(const v2f*)(tile + (wave * 16 + lid) * TS + k1);
            acc0 = wmma_f32_4(A0, B0, acc0);
            acc1 = wmma_f32_4(A1, B1, acc1);
        }
        __syncthreads();
    }
    v8f acc = acc0 + acc1;
    if (half == 0) {
        int c = cbase + wave * 16 + lid;
        #pragma unroll
        for (int r = 0; r < NHEADS; ++r)
            a[((size_t)b * NHEADS + r) * DMODEL + c] = acc[r];
    }
}

// ---------------------------------------------------------------------------
// K4: out[b, h*64+d] = sum_c a[b][h][c] * Wv[c, h*64+d] + bv[h*64+d]
// Per wave: 16 batch rows x 16 d cols, K=512.  All 16 M rows used.
// ---------------------------------------------------------------------------
__global__ void attn_proj(const float* __restrict__ a,
                          const float* __restrict__ Wkv,
                          const float* __restrict__ bkv,
                          float* __restrict__ out) {
    const int wave = threadIdx.x >> 5;
    const int lane = threadIdx.x & 31;
    const int half = lane >> 4;
    const int lid  = lane & 15;
    const int gw   = blockIdx.x * 8 + wave;      // 128 wave-tiles
    const int h    = gw >> 4;
    const int bt   = (gw >> 2) & 3;              // batch tile (16 rows)
    const int dt   = gw & 3;                     // d tile (16 cols)
    const int b0   = bt * 16;
    const int d0   = dt * 16;

    const float* arow = a + ((size_t)(b0 + lid) * NHEADS + h) * DMODEL;
    const float* wv   = Wkv + DMODEL + h * DHEAD + d0 + lid;
    v8f acc0 = {}, acc1 = {};
    #pragma unroll 4
    for (int c0 = 0; c0 < DMODEL; c0 += 8) {
        int k0 = c0 + 2 * half;
        int k1 = c0 + 4 + 2 * half;
        // A[m=batch, k=c] = a[b0+m][h][c] — consecutive K: single b64 load
        v2f A0 = *(const v2f*)(arow + k0);
        v2f A1 = *(const v2f*)(arow + k1);
        v2f B0, B1;
        // B[k=c, n=d] = Wv[c, h*64+d0+n]  (Wv = Wkv columns 512..1023)
        B0.x = wv[(size_t)(k0)     * (2 * DMODEL)];
        B0.y = wv[(size_t)(k0 + 1) * (2 * DMODEL)];
        B1.x = wv[(size_t)(k1)     * (2 * DMODEL)];
        B1.y = wv[(size_t)(k1 + 1) * (2 * DMODEL)];
        acc0 = wmma_f32_4(A0, B0, acc0);
        acc1 = wmma_f32_4(A1, B1, acc1);
    }
    v8f acc = acc0 + acc1;
    float bias = bkv[DMODEL + h * DHEAD + d0 + lid];
    #pragma unroll
    for (int r = 0; r < 8; ++r) {
        int brow = b0 + r + 8 * half;            // D layout: lanes16..31 = M+8
        out[(size_t)brow * DMODEL + h * DHEAD + d0 + lid] = acc[r] + bias;
    }
}

// ---------------------------------------------------------------------------
extern "C" void kernel_launch(void* const* d_in, const int* in_sizes, int n_in,
                              void* d_out, int out_size, void* d_ws, size_t ws_size,
                              hipStream_t stream) {
    const float* x   = (const float*)d_in[0];
    const float* q   = (const float*)d_in[1];
    const float* Wkv = (const float*)d_in[2];
    const float* bkv = (const float*)d_in[3];
    float* out = (float*)d_out;

    float* ws   = (float*)d_ws;
    float* wq16 = ws;                 //  8192 floats  [512][16], cols 8..15 = 0
    float* cq   = ws + 8192;          //  8192 floats  [8][1024]
    float* att  = ws + 16384;         // 64*8*1024 = 524288 floats (dots->attn)
    float* a    = ws + 540672;        // 64*8*512  = 262144 floats

    attn_prep<<<64, 256, 0, stream>>>(q, Wkv, bkv, wq16, cq);
    attn_dots<<<dim3(8, BATCH), 256, 0, stream>>>(x, wq16, cq, att);
    attn_softmax<<<BATCH * NHEADS, 256, 0, stream>>>(att);
    attn_av<<<dim3(4, BATCH), 256, 0, stream>>>(x, att, a);
    attn_proj<<<16, 256, 0, stream>>>(a, Wkv, bkv, out);
}